// EntropyBottleneck_16458314678740
// MI455X (gfx1250) — compile-verified
//
#include <hip/hip_runtime.h>
#include <math.h>

typedef float v2f __attribute__((ext_vector_type(2)));
typedef float v8f __attribute__((ext_vector_type(8)));

// ---------- math helpers ----------
__device__ __forceinline__ float softplusf(float x) {
  return (x > 20.f) ? x : log1pf(expf(x));
}

__device__ __forceinline__ float fast_tanh(float x) {
#if __has_builtin(__builtin_amdgcn_tanhf)
  return __builtin_amdgcn_tanhf(x);          // v_tanh_f32 (gfx1250 trans op)
#else
  float cx = fminf(fmaxf(x, -10.f), 10.f);
  float e = exp2f(cx * 2.885390081777927f);  // e^(2x)
  return (e - 1.f) / (e + 1.f);
#endif
}

__device__ __forceinline__ float sigmoidf_(float x) {
  return 1.f / (1.f + expf(-x));
}

// ---------- WMMA f32 16x16x4 : D = A(16x4) * B(4x16) + C ----------
// A layout (32-bit A 16x4): lanes 0-15 rows 0-15; vgpr0 = K0 (lanes<16) / K2 (lanes>=16),
//                           vgpr1 = K1 / K3.
// B layout (4x16):          vgpr j: lanes 0-15 = row K=j, lanes 16-31 = row K=j+2.
// C/D layout (16x16):       vgpr j: lanes 0-15 = row j, lanes 16-31 = row j+8.
__device__ __forceinline__ v8f wmma4_f32(v2f a, v2f b, v8f c) {
#if __has_builtin(__builtin_amdgcn_wmma_f32_16x16x4_f32)
  return __builtin_amdgcn_wmma_f32_16x16x4_f32(
      /*neg_a=*/false, a, /*neg_b=*/false, b,
      /*c_mod=*/(short)0, c, /*reuse_a=*/false, /*reuse_b=*/false);
#else
  // Shuffle-based emulation of the same layout (fallback only).
  int lane = threadIdx.x & 31;
  int half = lane >> 4;
  int n = lane & 15;
  float B0 = __shfl(b.x, n),      B1 = __shfl(b.y, n);
  float B2 = __shfl(b.x, n + 16), B3 = __shfl(b.y, n + 16);
  v8f d;
#pragma unroll
  for (int j = 0; j < 8; ++j) {
    int r = j + (half ? 8 : 0);
    float A0 = __shfl(a.x, r),      A1 = __shfl(a.y, r);
    float A2 = __shfl(a.x, r + 16), A3 = __shfl(a.y, r + 16);
    d[j] = fmaf(A0, B0, fmaf(A1, B1, fmaf(A2, B2, fmaf(A3, B3, c[j]))));
  }
  return d;
#endif
}

// ---------- per-layer, per-lane packed parameters ----------
struct Lay {
  float a0, a1;        // A-matrix entries for this lane (16x4 layout)
  float c0, c1, c2;    // bias in C/D layout (rows 0..2, lanes<16 only)
  float t0, t1, t2;    // tanh(f) per row (lanes<16 only)
};

// layer 0: M [C,3,1] -> A col 0 rows 0..2
__device__ __forceinline__ void setup_l0(const float* m, const float* b, const float* f,
                                         int ch, int lane, Lay& L) {
  int half = lane >> 4, row = lane & 15;
  const float* M = m + ch * 3;
  const float* B = b + ch * 3;
  const float* F = f + ch * 3;
  L.a0 = (!half && row < 3) ? softplusf(M[row]) : 0.f;
  L.a1 = 0.f;
  L.c0 = half ? 0.f : B[0];  L.c1 = half ? 0.f : B[1];  L.c2 = half ? 0.f : B[2];
  L.t0 = half ? 0.f : tanhf(F[0]);
  L.t1 = half ? 0.f : tanhf(F[1]);
  L.t2 = half ? 0.f : tanhf(F[2]);
}

// layers 1,2: M [C,3,3]
__device__ __forceinline__ void setup_l33(const float* m, const float* b, const float* f,
                                          int ch, int lane, Lay& L) {
  int half = lane >> 4, row = lane & 15;
  const float* M = m + ch * 9;
  const float* B = b + ch * 3;
  const float* F = f + ch * 3;
  float a0 = 0.f, a1 = 0.f;
  if (row < 3) {
    a0 = softplusf(M[row * 3 + (half ? 2 : 0)]);
    a1 = half ? 0.f : softplusf(M[row * 3 + 1]);
  }
  L.a0 = a0; L.a1 = a1;
  L.c0 = half ? 0.f : B[0];  L.c1 = half ? 0.f : B[1];  L.c2 = half ? 0.f : B[2];
  L.t0 = half ? 0.f : tanhf(F[0]);
  L.t1 = half ? 0.f : tanhf(F[1]);
  L.t2 = half ? 0.f : tanhf(F[2]);
}

// layer 3: M [C,1,3] -> A row 0
__device__ __forceinline__ void setup_l3(const float* m, const float* b, const float* f,
                                         int ch, int lane, Lay& L) {
  int half = lane >> 4, row = lane & 15;
  const float* M = m + ch * 3;
  L.a0 = (row == 0) ? softplusf(M[half ? 2 : 0]) : 0.f;
  L.a1 = (row == 0 && !half) ? softplusf(M[1]) : 0.f;
  L.c0 = half ? 0.f : b[ch];
  L.c1 = 0.f; L.c2 = 0.f;
  L.t0 = half ? 0.f : tanhf(f[ch]);
  L.t1 = 0.f; L.t2 = 0.f;
}

// Run the 4-layer chain; x0 is already in B-layout for layer 0 (lanes>=16: 0).
// Result is valid in lanes 0-15 (one point per lane).
__device__ __forceinline__ float eval_chain(Lay L0, Lay L1, Lay L2, Lay L3,
                                            float x0, int lane) {
  int half = lane >> 4;
  int nlo = lane & 15;

  v2f a, b;
  a.x = L0.a0; a.y = L0.a1;
  b.x = x0;    b.y = 0.f;
  v8f c = {L0.c0, L0.c1, L0.c2, 0.f, 0.f, 0.f, 0.f, 0.f};
  v8f d = wmma4_f32(a, b, c);
  float h0 = d[0] + L0.t0 * fast_tanh(d[0]);
  float h1 = d[1] + L0.t1 * fast_tanh(d[1]);
  float h2 = d[2] + L0.t2 * fast_tanh(d[2]);

  // layer 1
  float h2lo = __shfl(h2, nlo);                 // row2 -> K=2 slot (lanes 16-31)
  a.x = L1.a0; a.y = L1.a1;
  b.x = half ? h2lo : h0;  b.y = half ? 0.f : h1;
  v8f c1 = {L1.c0, L1.c1, L1.c2, 0.f, 0.f, 0.f, 0.f, 0.f};
  d = wmma4_f32(a, b, c1);
  h0 = d[0] + L1.t0 * fast_tanh(d[0]);
  h1 = d[1] + L1.t1 * fast_tanh(d[1]);
  h2 = d[2] + L1.t2 * fast_tanh(d[2]);

  // layer 2
  h2lo = __shfl(h2, nlo);
  a.x = L2.a0; a.y = L2.a1;
  b.x = half ? h2lo : h0;  b.y = half ? 0.f : h1;
  v8f c2 = {L2.c0, L2.c1, L2.c2, 0.f, 0.f, 0.f, 0.f, 0.f};
  d = wmma4_f32(a, b, c2);
  h0 = d[0] + L2.t0 * fast_tanh(d[0]);
  h1 = d[1] + L2.t1 * fast_tanh(d[1]);
  h2 = d[2] + L2.t2 * fast_tanh(d[2]);

  // layer 3
  h2lo = __shfl(h2, nlo);
  a.x = L3.a0; a.y = L3.a1;
  b.x = half ? h2lo : h0;  b.y = half ? 0.f : h1;
  v8f c3 = {L3.c0, 0.f, 0.f, 0.f, 0.f, 0.f, 0.f, 0.f};
  d = wmma4_f32(a, b, c3);
  float r = d[0];
  return r + L3.t0 * fast_tanh(r);
}

// ---------- main kernel ----------
// Block = 256 threads (8 wave32). Each tile = 16 points x 16 channels = 256 elems.
// Wave w owns channels {2w, 2w+1}; one WMMA chain handles 16 points of one channel.
__global__ void __launch_bounds__(256)
eb_kernel(const float* __restrict__ inputs, const float* __restrict__ noise,
          const float* __restrict__ m0, const float* __restrict__ b0, const float* __restrict__ f0,
          const float* __restrict__ m1, const float* __restrict__ b1, const float* __restrict__ f1,
          const float* __restrict__ m2, const float* __restrict__ b2, const float* __restrict__ f2,
          const float* __restrict__ m3, const float* __restrict__ b3, const float* __restrict__ f3,
          float* __restrict__ out, float* __restrict__ lik, int ntiles) {
  __shared__ float xs[256];  // x staged transposed: xs[c*16 + n]
  __shared__ float ls[256];  // likelihood tile:     ls[n*16 + c]

  const int tid  = threadIdx.x;
  const int lane = tid & 31;
  const int wave = tid >> 5;
  const int half = lane >> 4;
  const int row  = lane & 15;

  // One-time per-block parameter packing (softplus/tanh precomputed).
  Lay L0[2], L1[2], L2[2], L3[2];
#pragma unroll
  for (int cc = 0; cc < 2; ++cc) {
    int ch = wave * 2 + cc;
    setup_l0 (m0, b0, f0, ch, lane, L0[cc]);
    setup_l33(m1, b1, f1, ch, lane, L1[cc]);
    setup_l33(m2, b2, f2, ch, lane, L2[cc]);
    setup_l3 (m3, b3, f3, ch, lane, L3[cc]);
  }

  for (int t = blockIdx.x; t < ntiles; t += gridDim.x) {
    const int base = t * 256;

    // Phase 1: coalesced load, write outputs, stage x in LDS (transposed).
    float v = inputs[base + tid] + (noise[base + tid] - 0.5f);
    out[base + tid] = v;
    xs[(tid & 15) * 16 + (tid >> 4)] = v;
    __syncthreads();

    // Phase 2: WMMA chains (uniform control flow -> EXEC all ones).
#pragma unroll
    for (int cc = 0; cc < 2; ++cc) {
      int ch = wave * 2 + cc;
      float xv = xs[ch * 16 + row];                 // point n = row of this channel
      float xl = half ? 0.f : (xv - 0.5f);
      float xu = half ? 0.f : (xv + 0.5f);
      float lo = eval_chain(L0[cc], L1[cc], L2[cc], L3[cc], xl, lane);
      float up = eval_chain(L0[cc], L1[cc], L2[cc], L3[cc], xu, lane);
      if (half == 0) {
        float sum = lo + up;
        float s = (sum > 0.f) ? -1.f : ((sum < 0.f) ? 1.f : 0.f);
        float lk = fabsf(sigmoidf_(s * up) - sigmoidf_(s * lo));
        ls[row * 16 + ch] = fmaxf(lk, 1e-9f);
      }
    }
    __syncthreads();

    // Phase 3: coalesced likelihood store.
    lik[base + tid] = ls[tid];
    __syncthreads();
  }
}

extern "C" void kernel_launch(void* const* d_in, const int* in_sizes, int n_in,
                              void* d_out, int out_size, void* d_ws, size_t ws_size,
                              hipStream_t stream) {
  const float* inputs = (const float*)d_in[0];
  const float* noise  = (const float*)d_in[1];
  const float* m0 = (const float*)d_in[2];
  const float* b0 = (const float*)d_in[3];
  const float* f0 = (const float*)d_in[4];
  const float* m1 = (const float*)d_in[5];
  const float* b1 = (const float*)d_in[6];
  const float* f1 = (const float*)d_in[7];
  const float* m2 = (const float*)d_in[8];
  const float* b2 = (const float*)d_in[9];
  const float* f2 = (const float*)d_in[10];
  const float* m3 = (const float*)d_in[11];
  const float* b3 = (const float*)d_in[12];
  const float* f3 = (const float*)d_in[13];

  const int nelem  = in_sizes[0];     // N * C = 16777216
  const int ntiles = nelem / 256;     // 65536 tiles of 16 points x 16 channels
  float* out = (float*)d_out;
  float* lik = out + nelem;

  int blocks = ntiles < 2048 ? ntiles : 2048;
  eb_kernel<<<blocks, 256, 0, stream>>>(inputs, noise,
                                        m0, b0, f0, m1, b1, f1,
                                        m2, b2, f2, m3, b3, f3,
                                        out, lik, ntiles);
}